// HebbianConv2d_10720238370977
// MI455X (gfx1250) — compile-verified
//
#include <hip/hip_runtime.h>
#include <hip/hip_bf16.h>

// Conv2d 3x3, pad 1, fp32 via implicit GEMM on V_WMMA_F32_16X16X4_F32 (CDNA5).
// M = C_OUT = 128, N = B*H*W = 262144, K = C_IN*9 = 576.
// Activation tile staged into LDS via GLOBAL_LOAD_ASYNC_TO_LDS_B32 (ASYNCcnt).

typedef __attribute__((ext_vector_type(2))) float v2f;
typedef __attribute__((ext_vector_type(8))) float v8f;

#define CONST_B    16
#define CONST_CIN  64
#define CONST_H    128
#define CONST_W    128
#define CONST_COUT 128
#define CONST_K    576   // CIN * 9

// ---------------------------------------------------------------------------
// Repack weight [co][ci*9 + kk]  ->  wt[(k>>1)*256 + co*2 + (k&1)]
// where GEMM k = kk*64 + ci  (kk = kh*3+kw outer, ci inner).
// Makes the WMMA A-fragment a single coalesced 8-byte load per lane.
// ---------------------------------------------------------------------------
__global__ __launch_bounds__(256) void repack_weight_kernel(
    const float* __restrict__ w, float* __restrict__ wt) {
    int idx = blockIdx.x * 256 + threadIdx.x;       // over 576*128
    if (idx >= CONST_K * CONST_COUT) return;
    int co = idx & 127;
    int k  = idx >> 7;                              // GEMM k
    int kk = k >> 6;                                // kernel tap 0..8
    int ci = k & 63;                                // input channel
    float v = w[co * CONST_K + ci * 9 + kk];
    wt[(k >> 1) * 256 + co * 2 + (k & 1)] = v;
}

// ---------------------------------------------------------------------------
// Main conv kernel: one block per (b, h). Block tile: 128 (co) x 128 (w).
// 8 waves; wave m-strip = 16 co rows; 8 N-subtiles of 16 columns each.
// ---------------------------------------------------------------------------
__global__ __launch_bounds__(256) void conv3x3_wmma_f32_kernel(
    const float* __restrict__ x,     // [B][CIN][H][W]
    const float* __restrict__ wt,    // repacked weights, see above
    const float* __restrict__ bias,  // [COUT]
    float* __restrict__ out) {       // [B][COUT][H][W]

    const int bh = blockIdx.x;
    const int b  = bh >> 7;          // / H
    const int h  = bh & 127;         // % H

    // Activation tile with halo, channel-paired:
    // xs[pair][kh][c] = { x[b][2p][h+kh-1][c-1], x[b][2p+1][h+kh-1][c-1] }
    __shared__ v2f xs[32][3][130];

    const float* xb = x + (size_t)b * CONST_CIN * CONST_H * CONST_W;
    constexpr int HW = CONST_H * CONST_W;

    // ---- Zero the halo (ds stores, covered by DScnt / __syncthreads) ------
    {
        int idx = threadIdx.x;
        if (idx < 96) {                       // 32 pairs * 3 kh rows
            int pr = idx / 3, kh = idx % 3;
            xs[pr][kh][0]   = (v2f)(0.0f);
            xs[pr][kh][129] = (v2f)(0.0f);
        }
        if (h == 0 || h == 127) {             // one kh-plane is out of range
            const int khz = (h == 0) ? 0 : 2;
            for (int i = threadIdx.x; i < 32 * 130; i += 256) {
                int pr = i / 130, c = i % 130;
                xs[pr][khz][c] = (v2f)(0.0f);
            }
        }
    }

    // ---- Async global -> LDS copy of the interior (ASYNCcnt) --------------
    {
        const uint32_t xs_base = (uint32_t)(size_t)(void*)&xs[0][0][0];
        const int pty  = threadIdx.x >> 7;    // channel parity 0/1
        const int wcol = threadIdx.x & 127;   // output column
#pragma unroll
        for (int kh = 0; kh < 3; ++kh) {
            const int hh = h + kh - 1;
            if (hh >= 0 && hh < CONST_H) {
                unsigned long long g = (unsigned long long)(size_t)
                    (xb + (size_t)pty * HW + (size_t)hh * CONST_W + wcol);
                uint32_t l = xs_base +
                    (uint32_t)((kh * 130 + (wcol + 1)) * 8 + pty * 4);
#pragma unroll 8
                for (int i = 0; i < 32; ++i) {   // ci = 2*i + pty
                    asm volatile("global_load_async_to_lds_b32 %0, %1, off"
                                 :: "v"(l), "v"(g) : "memory");
                    g += (unsigned long long)2 * HW * sizeof(float);
                    l += 3 * 130 * 8;            // next channel-pair row
                }
            }
        }
    }
    // Async copies are NOT covered by __syncthreads(): drain ASYNCcnt first.
    asm volatile("s_wait_asynccnt 0x0" ::: "memory");
    __syncthreads();

    const int lane   = threadIdx.x & 31;
    const int wave   = threadIdx.x >> 5;
    const int m_base = wave * 16;        // this wave's co strip
    const int mlane  = lane & 15;        // M index (A) / N index (B,D)
    const int koff   = (lane >> 4) << 1; // K offset within quad: 0 or 2

    v8f acc[8];
#pragma unroll
    for (int nt = 0; nt < 8; ++nt) acc[nt] = (v8f)(0.0f);

    // K loop: 9 kernel taps x 16 ci-quads (K=4 each) = 144 WMMA K-steps.
    for (int kk = 0; kk < 9; ++kk) {
        const int kh = kk / 3;
        const int kw = kk % 3;
        const v2f* xrow_base = &xs[0][kh][0];
#pragma unroll 4
        for (int q = 0; q < 16; ++q) {
            const int kbase = kk * 64 + q * 4;   // GEMM k of quad start
            const int kA    = kbase + koff;      // this lane's K pair (even)
            // A fragment: wt[(kA>>1)*256 + co*2 + {0,1}] -> one b64 load.
            const v2f a = *(const v2f*)(wt + (size_t)(kA >> 1) * 256
                                           + (m_base + mlane) * 2);
            // B fragments from LDS: channel pair (q*4+koff)/2 of row kh.
            const v2f* xrow = xrow_base + (size_t)(q * 2 + (koff >> 1)) * 3 * 130;
#pragma unroll
            for (int nt = 0; nt < 8; ++nt) {
                const v2f bf = xrow[nt * 16 + mlane + kw];
                acc[nt] = __builtin_amdgcn_wmma_f32_16x16x4_f32(
                    /*neg_a=*/false, a, /*neg_b=*/false, bf,
                    /*c_mod=*/(short)0, acc[nt],
                    /*reuse_a=*/false, /*reuse_b=*/false);
            }
        }
    }

    // Store: D 16x16 layout -> VGPR r = row (m_base + mhi + r), N = lane&15.
    const int mhi = (lane >> 4) << 3;    // 0 or 8
    float* outb = out + (size_t)b * CONST_COUT * CONST_H * CONST_W
                      + h * CONST_W;
#pragma unroll
    for (int nt = 0; nt < 8; ++nt) {
        const int n = nt * 16 + mlane;
#pragma unroll
        for (int r = 0; r < 8; ++r) {
            const int co = m_base + mhi + r;
            outb[(size_t)co * CONST_H * CONST_W + n] = acc[nt][r] + bias[co];
        }
    }
}

// ---------------------------------------------------------------------------
// Launch
// ---------------------------------------------------------------------------
extern "C" void kernel_launch(void* const* d_in, const int* in_sizes, int n_in,
                              void* d_out, int out_size, void* d_ws, size_t ws_size,
                              hipStream_t stream) {
    const float* x    = (const float*)d_in[0];  // [16][64][128][128]
    const float* w    = (const float*)d_in[1];  // [128][576]
    const float* bias = (const float*)d_in[2];  // [128][1]
    float* out = (float*)d_out;                 // [16][128][128][128]
    float* wt  = (float*)d_ws;                  // needs 576*128*4 = 294912 B

    const int nW = CONST_K * CONST_COUT;
    repack_weight_kernel<<<(nW + 255) / 256, 256, 0, stream>>>(w, wt);

    conv3x3_wmma_f32_kernel<<<CONST_B * CONST_H, 256, 0, stream>>>(x, wt, bias, out);
}